// Softmax_43164421325263
// MI455X (gfx1250) — compile-verified
//
#include <hip/hip_runtime.h>
#include <hip/hip_bf16.h>
#include <stdint.h>

// Problem sizes (fixed by the reference).
#define BROWS   4096
#define CCOLS   32000
#define THREADS 320            // 10 wave32 waves; 8000 b128/row / 320 = 25 each
#define CHUNK   1280           // floats per chunk (THREADS * 4)
#define NCHUNK  25             // 25 * 1280 = 32000
#define NWAVES  (THREADS / 32)

typedef float v4f __attribute__((ext_vector_type(4)));  // native clang vector:
                                                        // valid for nontemporal builtins

// Online-softmax pair-combine: (m1,s1,t1) (+) (m2,s2,t2)
__device__ __forceinline__ void combine(float& m, float& s, float& t,
                                        float m2, float s2, float t2) {
    float nm = fmaxf(m, m2);
    float a  = __expf(m  - nm);
    float b  = __expf(m2 - nm);
    s = s * a + s2 * b;
    t = t * (a * a) + t2 * (b * b);
    m = nm;
}

__global__ __launch_bounds__(THREADS)
void ce_row_kernel(const float* __restrict__ logits,
                   const int*   __restrict__ labels,
                   float*       __restrict__ out_logits,
                   float*       __restrict__ ws) {
    __shared__ __align__(16) float sh[2][CHUNK];   // double-buffered stage, 10 KB
    __shared__ float redm[NWAVES], reds[NWAVES], redt[NWAVES];

    const int    tid   = threadIdx.x;
    const int    row   = blockIdx.x;
    const size_t rbase = (size_t)row * CCOLS;
    const float* gsrc  = logits     + rbase + (size_t)tid * 4;
    float*       gdst  = out_logits + rbase + (size_t)tid * 4;

    // LDS byte offsets of this thread's slot in each buffer.
    // (flat addr low 32 bits == LDS byte address on CDNA5; truncate)
    const unsigned lds0 = (unsigned)(uintptr_t)&sh[0][tid * 4];
    const unsigned lds1 = (unsigned)(uintptr_t)&sh[1][tid * 4];

    // ---- prologue: async-prefetch chunk 0 (HBM -> LDS, no VGPR round trip).
    // Touch-once stream: non-temporal so 1 GB of dead lines don't thrash L2.
    asm volatile("global_load_async_to_lds_b128 %0, %1, off th:TH_LOAD_NT"
                 :: "v"(lds0), "v"(gsrc) : "memory");

    float m = -3.402823466e38f, s = 0.0f, t = 0.0f;

    for (int c = 0; c < NCHUNK; ++c) {
        const int cur = c & 1;
        if (c + 1 < NCHUNK) {
            // prefetch next chunk into the other buffer
            const unsigned ldsn = cur ? lds0 : lds1;
            const float*   gn   = gsrc + (size_t)(c + 1) * CHUNK;
            asm volatile("global_load_async_to_lds_b128 %0, %1, off th:TH_LOAD_NT"
                         :: "v"(ldsn), "v"(gn) : "memory");
            // one older async load (chunk c) + one new -> wait until <=1
            asm volatile("s_wait_asynccnt 0x1" ::: "memory");
        } else {
            asm volatile("s_wait_asynccnt 0x0" ::: "memory");
        }

        // read back exactly the slot this thread prefetched (ds_load_b128)
        const v4f v = *(const v4f*)&sh[cur][tid * 4];

        // stream the logits copy out (b128 store, non-temporal: never re-read)
        __builtin_nontemporal_store(v, (v4f*)(gdst + (size_t)c * CHUNK));

        // online softmax accumulate over 4 values: 5 exps / 4 elems
        float mx = fmaxf(fmaxf(v.x, v.y), fmaxf(v.z, v.w));
        float nm = fmaxf(m, mx);
        float es = __expf(m - nm);
        float e0 = __expf(v.x - nm), e1 = __expf(v.y - nm);
        float e2 = __expf(v.z - nm), e3 = __expf(v.w - nm);
        s = s * es + ((e0 + e1) + (e2 + e3));
        t = t * (es * es) + ((e0 * e0 + e1 * e1) + (e2 * e2 + e3 * e3));
        m = nm;
    }

    // ---- wave32 shuffle-tree reduction of (m, s, t)
    for (int off = 16; off > 0; off >>= 1) {
        float m2 = __shfl_xor(m, off, 32);
        float s2 = __shfl_xor(s, off, 32);
        float t2 = __shfl_xor(t, off, 32);
        combine(m, s, t, m2, s2, t2);
    }
    const int wave = tid >> 5, lane = tid & 31;
    if (lane == 0) { redm[wave] = m; reds[wave] = s; redt[wave] = t; }
    __syncthreads();

    if (tid == 0) {
        float M = redm[0], S = reds[0], T = redt[0];
        for (int w = 1; w < NWAVES; ++w)       // fixed order -> deterministic
            combine(M, S, T, redm[w], reds[w], redt[w]);

        const int   lab = labels[row];
        const float xl  = logits[rbase + (size_t)lab];
        const float nll = (M + __logf(S)) - xl;

        // stats of softmax probs excluding the single max entry
        const float pmax    = 1.0f / S;        // exp(M-M)/S
        const float sq      = T / (S * S);     // sum p^2
        const float n       = (float)(CCOLS - 1);
        const float mean_ex = (1.0f - pmax) / n;
        const float var     = (sq - pmax * pmax - n * mean_ex * mean_ex) / (n - 1.0f);
        const float stdv    = sqrtf(fmaxf(var, 0.0f));

        ws[row]         = nll;
        ws[BROWS + row] = stdv;
    }
}

// Deterministic final mean over the 4096 per-row values.
__global__ __launch_bounds__(256)
void ce_finalize_kernel(const float* __restrict__ ws, float* __restrict__ out2) {
    __shared__ float ra[256], rb[256];
    const int tid = threadIdx.x;
    float a = 0.0f, b = 0.0f;
    for (int i = tid; i < BROWS; i += 256) {   // fixed strided order
        a += ws[i];
        b += ws[BROWS + i];
    }
    ra[tid] = a; rb[tid] = b;
    __syncthreads();
    for (int off = 128; off > 0; off >>= 1) {  // fixed tree order
        if (tid < off) { ra[tid] += ra[tid + off]; rb[tid] += rb[tid + off]; }
        __syncthreads();
    }
    if (tid == 0) {
        out2[0] = ra[0] * (1.0f / BROWS);      // loss (mean nll)
        out2[1] = rb[0] * (1.0f / BROWS);      // loss_nmpc (mean std)
    }
}

extern "C" void kernel_launch(void* const* d_in, const int* in_sizes, int n_in,
                              void* d_out, int out_size, void* d_ws, size_t ws_size,
                              hipStream_t stream) {
    const float* logits = (const float*)d_in[0];
    const int*   labels = (const int*)d_in[1];   // jax default: int64 -> int32
    float*       out    = (float*)d_out;         // [B*C logits copy][loss][loss_nmpc]
    float*       ws     = (float*)d_ws;          // 2*B floats of scratch

    ce_row_kernel<<<BROWS, THREADS, 0, stream>>>(logits, labels, out, ws);
    ce_finalize_kernel<<<1, 256, 0, stream>>>(ws, out + (size_t)BROWS * CCOLS);
}